// TransSolver_73478300500051
// MI455X (gfx1250) — compile-verified
//
#include <hip/hip_runtime.h>

#define TFRAMES 131072
#define CHUNK   256
#define NBLK    (TFRAMES / CHUNK)   /* 512 */
#define FLOORY  (-0.93f)
#define GRAV    (-0.0018f)
#define NEG_INF (-3.0e38f)

typedef __attribute__((ext_vector_type(2))) float v2f;
typedef __attribute__((ext_vector_type(8))) float v8f;

// ---------------------------------------------------------------------------
// Pass 0: pose passthrough (bandwidth-bound: ~226 MB total traffic)
// ---------------------------------------------------------------------------
__global__ __launch_bounds__(256) void copy_pose_kernel(const float4* __restrict__ src,
                                                        float4* __restrict__ dst, int n4) {
  int stride = gridDim.x * blockDim.x;
  for (int i = blockIdx.x * blockDim.x + threadIdx.x; i < n4; i += stride) {
    __builtin_prefetch(&src[i + 2 * stride], 0, 0);   // global_prefetch_b8
    dst[i] = src[i];
  }
}

// ---------------------------------------------------------------------------
// Pass 1: forward kinematics per frame. Two rolling chains:
//   A: locals {1,3,5,7} (pose joints {1,4,7,10}) -> p5, p7
//   B: locals {2,4,6,8} (pose joints {2,5,8,11}) -> p6, p8
// feet order [7,5,8,6]; fmin = min_y(p5..p8)
// ---------------------------------------------------------------------------
__device__ __forceinline__ void mat3_mul(const float* A, const float* B, float* D) {
#pragma unroll
  for (int r = 0; r < 3; ++r) {
#pragma unroll
    for (int c = 0; c < 3; ++c)
      D[r*3+c] = __builtin_fmaf(A[r*3+0], B[0+c],
                 __builtin_fmaf(A[r*3+1], B[3+c], A[r*3+2] * B[6+c]));
  }
}

__global__ __launch_bounds__(256) void fk_kernel(const float* __restrict__ pose,
                                                 const float* __restrict__ bone,
                                                 float* __restrict__ feet,
                                                 float* __restrict__ fmin) {
  int t = blockIdx.x * blockDim.x + threadIdx.x;
  if (t >= TFRAMES) return;
  const float* fp = pose + (size_t)t * 216;

  float Rg0[9];
#pragma unroll
  for (int e = 0; e < 9; ++e) Rg0[e] = fp[e];          // pose joint 0
  float p0x = bone[0], p0y = bone[1], p0z = bone[2];

  float p5[3], p6[3], p7[3], p8[3];

  // ---- chain A ----
  {
    float Rg[9], Rn[9], Rl[9];
    float px = p0x, py = p0y, pz = p0z;
#pragma unroll
    for (int e = 0; e < 9; ++e) Rg[e] = Rg0[e];
    const int locals[4] = {1, 3, 5, 7};
    const int pjoint[4] = {1, 4, 7, 10};
#pragma unroll
    for (int s = 0; s < 4; ++s) {
      int li = locals[s];
      float bx = bone[li*3+0], by = bone[li*3+1], bz = bone[li*3+2];
      float qx = px + Rg[0]*bx + Rg[1]*by + Rg[2]*bz;
      float qy = py + Rg[3]*bx + Rg[4]*by + Rg[5]*bz;
      float qz = pz + Rg[6]*bx + Rg[7]*by + Rg[8]*bz;
      px = qx; py = qy; pz = qz;
      if (li == 5) { p5[0]=qx; p5[1]=qy; p5[2]=qz; }
      if (li == 7) { p7[0]=qx; p7[1]=qy; p7[2]=qz; }
      if (s < 3) {
#pragma unroll
        for (int e = 0; e < 9; ++e) Rl[e] = fp[pjoint[s]*9 + e];
        mat3_mul(Rg, Rl, Rn);
#pragma unroll
        for (int e = 0; e < 9; ++e) Rg[e] = Rn[e];
      }
    }
  }
  // ---- chain B ----
  {
    float Rg[9], Rn[9], Rl[9];
    float px = p0x, py = p0y, pz = p0z;
#pragma unroll
    for (int e = 0; e < 9; ++e) Rg[e] = Rg0[e];
    const int locals[4] = {2, 4, 6, 8};
    const int pjoint[4] = {2, 5, 8, 11};
#pragma unroll
    for (int s = 0; s < 4; ++s) {
      int li = locals[s];
      float bx = bone[li*3+0], by = bone[li*3+1], bz = bone[li*3+2];
      float qx = px + Rg[0]*bx + Rg[1]*by + Rg[2]*bz;
      float qy = py + Rg[3]*bx + Rg[4]*by + Rg[5]*bz;
      float qz = pz + Rg[6]*bx + Rg[7]*by + Rg[8]*bz;
      px = qx; py = qy; pz = qz;
      if (li == 6) { p6[0]=qx; p6[1]=qy; p6[2]=qz; }
      if (li == 8) { p8[0]=qx; p8[1]=qy; p8[2]=qz; }
      if (s < 3) {
#pragma unroll
        for (int e = 0; e < 9; ++e) Rl[e] = fp[pjoint[s]*9 + e];
        mat3_mul(Rg, Rl, Rn);
#pragma unroll
        for (int e = 0; e < 9; ++e) Rg[e] = Rn[e];
      }
    }
  }

  float* ft = feet + (size_t)t * 12;           // FEET order [7,5,8,6]
  ft[0]=p7[0]; ft[1] =p7[1]; ft[2] =p7[2];
  ft[3]=p5[0]; ft[4] =p5[1]; ft[5] =p5[2];
  ft[6]=p8[0]; ft[7] =p8[1]; ft[8] =p8[2];
  ft[9]=p6[0]; ft[10]=p6[1]; ft[11]=p6[2];
  fmin[t] = fminf(fminf(p5[1], p6[1]), fminf(p7[1], p8[1]));
}

// ---------------------------------------------------------------------------
// Pass 2: per-frame velocity (select foot by index) + floor constant
// ---------------------------------------------------------------------------
__global__ __launch_bounds__(256) void vel_kernel(const float* __restrict__ feet,
                                                  const float* __restrict__ fmin,
                                                  const int* __restrict__ index,
                                                  float* __restrict__ VX, float* __restrict__ VY,
                                                  float* __restrict__ VZ, float* __restrict__ CC) {
  int t = blockIdx.x * blockDim.x + threadIdx.x;
  if (t >= TFRAMES) return;
  int idx = index[t];
  float vx = 0.f, vy = 0.f, vz = 0.f;
  if (t > 0) {
    const float* fprev = feet + (size_t)(t - 1) * 12 + idx * 3;
    const float* fcur  = feet + (size_t)t * 12 + idx * 3;
    vx = fprev[0] - fcur[0];
    vy = fprev[1] - fcur[1];
    vz = fprev[2] - fcur[2];
  }
  VX[t] = vx;
  VY[t] = vy + GRAV;
  VZ[t] = vz;
  CC[t] = FLOORY - fmin[t];
}

// ---------------------------------------------------------------------------
// Scan algebra for y: f(r) = max(r + S, M); compose(a,b) = (aS+bS, max(aM+bS, bM))
// Pass 3: per-wave (256-frame chunk) aggregates via ordered Hillis-Steele scan
// ---------------------------------------------------------------------------
__global__ __launch_bounds__(32) void scan_partial_kernel(const float* __restrict__ VX,
                                                          const float* __restrict__ VY,
                                                          const float* __restrict__ VZ,
                                                          const float* __restrict__ CC,
                                                          float* __restrict__ BAGG) {
  int b = blockIdx.x, lane = threadIdx.x;
  int base = b * CHUNK + lane * 8;
  float sx = 0.f, sz = 0.f, S = 0.f, M = NEG_INF;
#pragma unroll
  for (int i = 0; i < 8; ++i) {
    int t = base + i;
    sx += VX[t];
    sz += VZ[t];
    float vy = VY[t];
    S += vy;
    M = fmaxf(M + vy, CC[t]);
  }
#pragma unroll
  for (int d = 1; d < 32; d <<= 1) {       // order-preserving inclusive scan
    float px = __shfl_up(sx, d, 32);
    float pz = __shfl_up(sz, d, 32);
    float pS = __shfl_up(S, d, 32);
    float pM = __shfl_up(M, d, 32);
    if (lane >= d) { M = fmaxf(pM + S, M); S = pS + S; sx += px; sz += pz; }
  }
  if (lane == 31) {
    BAGG[b*4+0] = sx; BAGG[b*4+1] = sz; BAGG[b*4+2] = S; BAGG[b*4+3] = M;
  }
}

// ---------------------------------------------------------------------------
// Pass 4: single-block scan of the 512 chunk aggregates -> per-chunk prefixes
// ---------------------------------------------------------------------------
__global__ __launch_bounds__(NBLK) void scan_blocks_kernel(const float* __restrict__ BAGG,
                                                           float* __restrict__ BPRE) {
  __shared__ float sx[NBLK], sz[NBLK], sS[NBLK], sM[NBLK];
  int tid = threadIdx.x;
  float x = BAGG[tid*4+0], z = BAGG[tid*4+1], S = BAGG[tid*4+2], M = BAGG[tid*4+3];
  sx[tid] = x; sz[tid] = z; sS[tid] = S; sM[tid] = M;
  __syncthreads();
  for (int d = 1; d < NBLK; d <<= 1) {
    float px = 0.f, pz = 0.f, pS = 0.f, pM = NEG_INF;
    if (tid >= d) { px = sx[tid-d]; pz = sz[tid-d]; pS = sS[tid-d]; pM = sM[tid-d]; }
    __syncthreads();
    if (tid >= d) { M = fmaxf(pM + S, M); S = pS + S; x += px; z += pz; }
    sx[tid] = x; sz[tid] = z; sS[tid] = S; sM[tid] = M;
    __syncthreads();
  }
  float ex = 0.f, ez = 0.f, eS = 0.f, eM = NEG_INF;
  if (tid > 0) { ex = sx[tid-1]; ez = sz[tid-1]; eS = sS[tid-1]; eM = sM[tid-1]; }
  BPRE[tid*3+0] = ex;
  BPRE[tid*3+1] = ez;
  BPRE[tid*3+2] = fmaxf(eS, eM);   // r entering the chunk, starting from r=0
}

// ---------------------------------------------------------------------------
// Pass 5: apply. x/z: in-segment (16-frame) inclusive cumsum as L(16x16 ones,
// lower-tri) @ X via 4 chained V_WMMA_F32_16X16X4_F32; y: max-plus in VALU.
// One wave per 256-frame chunk; all 32 lanes active (WMMA EXEC requirement).
// ---------------------------------------------------------------------------
__device__ __forceinline__ v8f wmma_cumsum16(const float* __restrict__ src,
                                             int base, int half, int ln) {
  v8f acc = {0.f, 0.f, 0.f, 0.f, 0.f, 0.f, 0.f, 0.f};
#pragma unroll
  for (int j = 0; j < 4; ++j) {
    int k0 = 4 * j + 2 * half;        // K slice held by this half-wave
    v2f a, bv;
    a.x = (ln >= k0)     ? 1.0f : 0.0f;   // L[m][k0]
    a.y = (ln >= k0 + 1) ? 1.0f : 0.0f;   // L[m][k0+1]
    bv.x = src[base + ln * 16 + k0];      // X[k0][n],   n = ln
    bv.y = src[base + ln * 16 + k0 + 1];  // X[k0+1][n]
    acc = __builtin_amdgcn_wmma_f32_16x16x4_f32(false, a, false, bv,
                                                (short)0, acc, false, false);
  }
  return acc;
}

__device__ __forceinline__ void store_cumsum(v8f acc, float choff, int base,
                                             int half, int ln,
                                             float* __restrict__ trans, int comp) {
  // column-n total (m=15) lives in VGPR7 of lanes 16..31
  float tot = acc[7];
  float segoff = 0.f;
#pragma unroll
  for (int k = 0; k < 16; ++k) {
    float tk = __shfl(tot, 16 + k, 32);
    if (k < ln) segoff += tk;             // exclusive scan of segment totals
  }
  float add = choff + segoff;
#pragma unroll
  for (int r = 0; r < 8; ++r) {
    int frame = base + ln * 16 + r + 8 * half;   // D[m][n], m = r + 8*half
    trans[frame * 3 + comp] = acc[r] + add;
  }
}

__global__ __launch_bounds__(32) void trans_apply_kernel(const float* __restrict__ VX,
                                                         const float* __restrict__ VY,
                                                         const float* __restrict__ VZ,
                                                         const float* __restrict__ CC,
                                                         const float* __restrict__ BPRE,
                                                         float* __restrict__ trans) {
  int b = blockIdx.x, lane = threadIdx.x;
  int base = b * CHUNK;
  float offx = BPRE[b*3+0], offz = BPRE[b*3+1], r_in = BPRE[b*3+2];
  int half = lane >> 4, ln = lane & 15;

  // ---- x / z channels via WMMA matmul-scan ----
  v8f ax = wmma_cumsum16(VX, base, half, ln);
  store_cumsum(ax, offx, base, half, ln, trans, 0);
  v8f az = wmma_cumsum16(VZ, base, half, ln);
  store_cumsum(az, offz, base, half, ln, trans, 2);

  // ---- y channel: max-plus scan ----
  int ybase = base + lane * 8;
  float vy[8], cc[8];
  float S = 0.f, M = NEG_INF;
#pragma unroll
  for (int i = 0; i < 8; ++i) {
    vy[i] = VY[ybase + i];
    cc[i] = CC[ybase + i];
    S += vy[i];
    M = fmaxf(M + vy[i], cc[i]);
  }
#pragma unroll
  for (int d = 1; d < 32; d <<= 1) {
    float pS = __shfl_up(S, d, 32);
    float pM = __shfl_up(M, d, 32);
    if (lane >= d) { M = fmaxf(pM + S, M); S = pS + S; }
  }
  float eS = __shfl_up(S, 1, 32);
  float eM = __shfl_up(M, 1, 32);
  if (lane == 0) { eS = 0.f; eM = NEG_INF; }
  float r = fmaxf(r_in + eS, eM);
#pragma unroll
  for (int i = 0; i < 8; ++i) {
    r = fmaxf(r + vy[i], cc[i]);
    trans[(ybase + i) * 3 + 1] = r;
  }
}

// ---------------------------------------------------------------------------
extern "C" void kernel_launch(void* const* d_in, const int* in_sizes, int n_in,
                              void* d_out, int out_size, void* d_ws, size_t ws_size,
                              hipStream_t stream) {
  const float* pose  = (const float*)d_in[0];
  const float* bone  = (const float*)d_in[1];
  const int*   index = (const int*)d_in[2];
  float* out = (float*)d_out;
  float* ws  = (float*)d_ws;

  const size_t T = TFRAMES;
  float* FEET = ws;                       // T*12
  float* FMIN = ws + 12 * T;              // T
  float* VX   = ws + 13 * T;              // T
  float* VY   = ws + 14 * T;              // T
  float* VZ   = ws + 15 * T;              // T
  float* CC   = ws + 16 * T;              // T
  float* BAGG = ws + 17 * T;              // NBLK*4
  float* BPRE = BAGG + 4 * NBLK;          // NBLK*3
  float* trans = out + T * 216;           // output: pose (T*216) then trans (T*3)

  int n4 = (int)(T * 216 / 4);
  copy_pose_kernel<<<2048, 256, 0, stream>>>((const float4*)pose, (float4*)out, n4);
  fk_kernel<<<TFRAMES / 256, 256, 0, stream>>>(pose, bone, FEET, FMIN);
  vel_kernel<<<TFRAMES / 256, 256, 0, stream>>>(FEET, FMIN, index, VX, VY, VZ, CC);
  scan_partial_kernel<<<NBLK, 32, 0, stream>>>(VX, VY, VZ, CC, BAGG);
  scan_blocks_kernel<<<1, NBLK, 0, stream>>>(BAGG, BPRE);
  trans_apply_kernel<<<NBLK, 32, 0, stream>>>(VX, VY, VZ, CC, BPRE, trans);
}